// GCN_3ly_66709432041783
// MI455X (gfx1250) — compile-verified
//
#include <hip/hip_runtime.h>
#include <hip/hip_bf16.h>

typedef __attribute__((ext_vector_type(2))) float v2f;
typedef __attribute__((ext_vector_type(8))) float v8f;

// ---------------------------------------------------------------------------
// Degree kernels: deg[i] starts at 1.0 (self loop), +1 per incoming edge,
// then dinv = rsqrt(deg) in place.
// ---------------------------------------------------------------------------
__global__ void gcn_deg_init(float* __restrict__ deg, int n) {
    int i = blockIdx.x * blockDim.x + threadIdx.x;
    if (i < n) deg[i] = 1.0f;
}

__global__ void gcn_deg_count(float* __restrict__ deg,
                              const int* __restrict__ dst, int nEdges) {
    int e = blockIdx.x * blockDim.x + threadIdx.x;
    if (e < nEdges) unsafeAtomicAdd(&deg[dst[e]], 1.0f);
}

__global__ void gcn_deg_finalize(float* __restrict__ deg, int n) {
    int i = blockIdx.x * blockDim.x + threadIdx.x;
    if (i < n) {
        float d = deg[i];
        deg[i] = (d > 0.0f) ? rsqrtf(d) : 0.0f;
    }
}

// ---------------------------------------------------------------------------
// Dense transform: Out[nRows x outDim] = A[nRows x inDim] * W[inDim x outDim]
// One wave32 computes one 16x16 tile via V_WMMA_F32_16X16X4_F32.
// Requires nRows % 16 == 0, inDim % 4 == 0, outDim % 16 == 0.
// ---------------------------------------------------------------------------
__global__ void gcn_gemm_wmma(const float* __restrict__ A,
                              const float* __restrict__ W,
                              float* __restrict__ Out,
                              int nRows, int inDim, int outDim) {
    const int wavesPerBlock = blockDim.x >> 5;
    const int waveId = threadIdx.x >> 5;
    const int lane   = threadIdx.x & 31;
    const int nTiles = outDim >> 4;
    const int totalTiles = (nRows >> 4) * nTiles;
    const int tile = blockIdx.x * wavesPerBlock + waveId;
    if (tile >= totalTiles) return;            // wave-uniform: EXEC stays full

    const int mBase = (tile / nTiles) << 4;
    const int nBase = (tile % nTiles) << 4;
    const int hi  = lane >> 4;                 // 0: lanes 0-15, 1: lanes 16-31
    const int l16 = lane & 15;

    const float* Arow = A + (size_t)(mBase + l16) * inDim + 2 * hi;
    const float* Bcol = W + nBase + l16;

    v8f acc = {};
    for (int k = 0; k < inDim; k += 4) {
        // A 16x4 frag: lane L, vgpr v <-> A[L%16][2*(L/16)+v]
        v2f a;
        a.x = Arow[k];
        a.y = Arow[k + 1];
        // B 4x16 frag: lane L, vgpr v <-> B[2*(L/16)+v][L%16]
        const int kb = k + 2 * hi;
        v2f b;
        b.x = Bcol[(size_t)(kb)     * outDim];
        b.y = Bcol[(size_t)(kb + 1) * outDim];
        acc = __builtin_amdgcn_wmma_f32_16x16x4_f32(
            false, a, false, b, (short)0, acc, false, false);
    }

    // D 16x16: lane L, vgpr v <-> D[v + 8*(L/16)][L%16]
    float* orow = Out + (size_t)(mBase + 8 * hi) * outDim + nBase + l16;
#pragma unroll
    for (int v = 0; v < 8; ++v) orow[(size_t)v * outDim] = acc[v];
}

// ---------------------------------------------------------------------------
// agg[i][f] = bias[f] + dinv[i]^2 * h[i][f]   (self-loop term + bias init)
// dim must be a power of two (dimShift = log2(dim)).
// ---------------------------------------------------------------------------
__global__ void gcn_agg_init(const float* __restrict__ h,
                             float* __restrict__ agg,
                             const float* __restrict__ dinv,
                             const float* __restrict__ bias,
                             int nNodes, int dimShift) {
    long long tid = (long long)blockIdx.x * blockDim.x + threadIdx.x;
    long long total = (long long)nNodes << dimShift;
    if (tid >= total) return;
    int node = (int)(tid >> dimShift);
    int f    = (int)(tid & ((1 << dimShift) - 1));
    float di = dinv[node];
    agg[tid] = bias[f] + di * di * h[tid];
}

// ---------------------------------------------------------------------------
// Edge aggregation: agg[dst] += dinv[src]*dinv[dst]*h[src]
// (dim/4) lanes cooperate per edge -> coalesced float4 gathers, fp32 atomics.
// ---------------------------------------------------------------------------
__global__ void gcn_aggregate(const float* __restrict__ h,
                              float* __restrict__ agg,
                              const int* __restrict__ src,
                              const int* __restrict__ dst,
                              const float* __restrict__ dinv,
                              int nEdges, int dimShift) {
    const int chunkShift = dimShift - 2;                 // float4 chunks
    long long tid = (long long)blockIdx.x * blockDim.x + threadIdx.x;
    long long e = tid >> chunkShift;
    int c = (int)(tid & ((1 << chunkShift) - 1));
    if (e >= nEdges) return;
    int s = src[e];
    int d = dst[e];
    float w = dinv[s] * dinv[d];
    const float4 v =
        *reinterpret_cast<const float4*>(h + ((size_t)s << dimShift) + 4 * c);
    float* p = agg + ((size_t)d << dimShift) + 4 * c;
    unsafeAtomicAdd(p + 0, v.x * w);
    unsafeAtomicAdd(p + 1, v.y * w);
    unsafeAtomicAdd(p + 2, v.z * w);
    unsafeAtomicAdd(p + 3, v.w * w);
}

__global__ void gcn_relu(float* __restrict__ x, long long n) {
    long long i = (long long)blockIdx.x * blockDim.x + threadIdx.x;
    if (i < n) x[i] = fmaxf(x[i], 0.0f);
}

// ---------------------------------------------------------------------------
// out[i] = sigmoid( relu(agg[i][0:32]) . Wl + bl )
// ---------------------------------------------------------------------------
__global__ void gcn_final(const float* __restrict__ agg,
                          const float* __restrict__ Wl,
                          const float* __restrict__ bl,
                          float* __restrict__ out, int nNodes) {
    int i = blockIdx.x * blockDim.x + threadIdx.x;
    if (i >= nNodes) return;
    const float4* row = reinterpret_cast<const float4*>(agg + (size_t)i * 32);
    float s = bl[0];
#pragma unroll
    for (int c = 0; c < 8; ++c) {
        float4 v = row[c];
        float4 w = reinterpret_cast<const float4*>(Wl)[c];
        s += fmaxf(v.x, 0.0f) * w.x + fmaxf(v.y, 0.0f) * w.y +
             fmaxf(v.z, 0.0f) * w.z + fmaxf(v.w, 0.0f) * w.w;
    }
    out[i] = 1.0f / (1.0f + expf(-s));
}

// ---------------------------------------------------------------------------
// Host-side launch
// ---------------------------------------------------------------------------
extern "C" void kernel_launch(void* const* d_in, const int* in_sizes, int n_in,
                              void* d_out, int out_size, void* d_ws, size_t ws_size,
                              hipStream_t stream) {
    const float* x   = (const float*)d_in[0];
    const int*   ei  = (const int*)d_in[1];      // [2, E]; integer -> int32
    const float* W1  = (const float*)d_in[2];
    const float* b1  = (const float*)d_in[3];
    const float* W2  = (const float*)d_in[4];
    const float* b2  = (const float*)d_in[5];
    const float* W3  = (const float*)d_in[6];
    const float* b3  = (const float*)d_in[7];
    const float* Wl  = (const float*)d_in[8];
    const float* bl  = (const float*)d_in[9];
    float* out = (float*)d_out;

    const int IN_DIM = 128, H1 = 128, H2 = 64, H3 = 32;
    const int N = in_sizes[0] / IN_DIM;
    const int E = in_sizes[1] / 2;
    const int* srcIdx = ei;
    const int* dstIdx = ei + E;

    float* dinv = (float*)d_ws;                  // [N]
    float* h    = dinv + N;                      // [N x 128] (reused per layer)
    float* agg  = h + (size_t)N * 128;           // [N x 128] (reused per layer)

    const int B = 256;
    auto cdiv = [](long long a, long long b) { return (int)((a + b - 1) / b); };

    // --- degrees -> dinv ---
    gcn_deg_init<<<cdiv(N, B), B, 0, stream>>>(dinv, N);
    gcn_deg_count<<<cdiv(E, B), B, 0, stream>>>(dinv, dstIdx, E);
    gcn_deg_finalize<<<cdiv(N, B), B, 0, stream>>>(dinv, N);

    // Per-layer helper sequence (all stream-ordered):
    //   h = in @ W ; agg = b + dinv^2*h ; agg += edges ; relu(agg)
    struct Layer { const float* W; const float* b; int din; int dout; int shift; };
    const Layer layers[3] = {
        { W1, b1, IN_DIM, H1, 7 },
        { W2, b2, H1,     H2, 6 },
        { W3, b3, H2,     H3, 5 },
    };

    const float* cur = x;
    for (int l = 0; l < 3; ++l) {
        const Layer& L = layers[l];
        const int tiles = (N / 16) * (L.dout / 16);
        gcn_gemm_wmma<<<cdiv(tiles, 4), 128, 0, stream>>>(cur, L.W, h, N, L.din, L.dout);
        gcn_agg_init<<<cdiv((long long)N * L.dout, B), B, 0, stream>>>(h, agg, dinv, L.b, N, L.shift);
        gcn_aggregate<<<cdiv((long long)E << (L.shift - 2), B), B, 0, stream>>>(
            h, agg, srcIdx, dstIdx, dinv, E, L.shift);
        if (l < 2)  // layer-3 ReLU is folded into gcn_final
            gcn_relu<<<cdiv((long long)N * L.dout, B), B, 0, stream>>>(agg, (long long)N * L.dout);
        cur = agg;
    }

    gcn_final<<<cdiv(N, B), B, 0, stream>>>(agg, Wl, bl, out, N);
}